// VectorQuantizer_23252952941094
// MI455X (gfx1250) — compile-verified
//
#include <hip/hip_runtime.h>
#include <math.h>

typedef __attribute__((ext_vector_type(2))) float v2f;
typedef __attribute__((ext_vector_type(8))) float v8f;

#define N_E   1024
#define E_DIM 128
#define NROWS 16384
#define BETA  0.25f

// d_out layout (floats), tuple order: loss, z_q_st, perplexity, min_encodings, indices
#define OFF_LOSS 0
#define OFF_ZQ   1
#define OFF_PERP 2097153          // 1 + 32*512*128
#define OFF_ENC  2097154
#define OFF_IDX  18874370         // OFF_ENC + 16384*1024

// ---------------- kernel 1: zero one-hot region, counts, scalar slots -------
__global__ void vq_zero_kernel(float* __restrict__ out, float* __restrict__ counts) {
    size_t i      = (size_t)blockIdx.x * blockDim.x + threadIdx.x;
    size_t stride = (size_t)gridDim.x * blockDim.x;
    float2* enc = (float2*)(out + OFF_ENC);          // OFF_ENC even -> 8B aligned
    const size_t n2 = (size_t)N_E * NROWS / 2;
    for (size_t j = i; j < n2; j += stride) enc[j] = make_float2(0.f, 0.f);
    if (i < N_E) counts[i] = 0.f;
    if (i == 0) { out[OFF_LOSS] = 0.f; out[OFF_PERP] = 0.f; }
}

// ---------------- kernel 2: ||e||^2 per code --------------------------------
__global__ void vq_e2_kernel(const float* __restrict__ emb, float* __restrict__ e2) {
    int i = blockIdx.x * blockDim.x + threadIdx.x;
    if (i < N_E) {
        const float* p = emb + (size_t)i * E_DIM;
        float s = 0.f;
        #pragma unroll 8
        for (int c = 0; c < E_DIM; ++c) s += p[c] * p[c];
        e2[i] = s;
    }
}

// ---------------- kernel 3: WMMA f32 argmin ---------------------------------
// One wave32 per 16-row tile of z. D = A(16x4) x B(4x16) accumulated over K=128.
__global__ void __launch_bounds__(256)
vq_argmin_wmma(const float* __restrict__ z, const float* __restrict__ emb,
               const float* __restrict__ e2, int* __restrict__ out_idx) {
    const int lane = threadIdx.x & 31;
    const int wave = threadIdx.x >> 5;
    const int tile = blockIdx.x * 8 + wave;     // 0..1023 (16-row tiles)
    const int m    = lane & 15;                 // A: row M / B: col N / e2 index
    const int half = lane >> 4;                 // K sub-slot selector

    // Preload all A fragments for this row tile: 32 K-steps x 2 f32 per lane.
    const float* zrow = z + ((size_t)tile * 16 + m) * E_DIM;
    v2f a[32];
    #pragma unroll
    for (int s = 0; s < 32; ++s)
        a[s] = *(const v2f*)(zrow + 4 * s + 2 * half);

    float bestv[8];
    int   besti[8];
    #pragma unroll
    for (int v = 0; v < 8; ++v) { bestv[v] = 3.4e38f; besti[v] = 0; }

    for (int nt = 0; nt < 64; ++nt) {           // 64 tiles of 16 codes
        const float* erow = emb + ((size_t)nt * 16 + m) * E_DIM;
        const float  e2l  = e2[nt * 16 + m];    // e2 of code N (= lane's m)
        v8f acc = {};
        #pragma unroll
        for (int s = 0; s < 32; ++s) {
            v2f b = *(const v2f*)(erow + 4 * s + 2 * half);
            acc = __builtin_amdgcn_wmma_f32_16x16x4_f32(
                false, a[s], false, b, (short)0, acc, false, false);
        }
        // lane holds D[M=v+8*half][N=m]; score = ||e_N||^2 - 2 * z_M . e_N
        #pragma unroll
        for (int v = 0; v < 8; ++v) {
            float score = e2l - 2.0f * acc[v];
            if (score < bestv[v]) { bestv[v] = score; besti[v] = nt * 16 + m; }
        }
    }

    // Reduce over the 16 lanes sharing the same M (first-index tie-break).
    #pragma unroll
    for (int v = 0; v < 8; ++v) {
        float bv = bestv[v];
        int   bi = besti[v];
        #pragma unroll
        for (int off = 8; off > 0; off >>= 1) {
            float ov = __shfl_xor(bv, off, 16);
            int   oi = __shfl_xor(bi, off, 16);
            if (ov < bv || (ov == bv && oi < bi)) { bv = ov; bi = oi; }
        }
        if (m == 0) out_idx[tile * 16 + v + 8 * half] = bi;
    }
}

// ---------------- kernel 4: gather z_q, one-hot, loss, counts ---------------
__global__ void __launch_bounds__(256)
vq_gather_kernel(const float* __restrict__ z, const float* __restrict__ emb,
                 const int* __restrict__ idx, float* __restrict__ out,
                 float* __restrict__ counts) {
    const int lane = threadIdx.x & 31;
    const int r    = (blockIdx.x * blockDim.x + threadIdx.x) >> 5; // one wave/row
    const int id   = idx[r];
    const float* e  = emb + (size_t)id * E_DIM;
    const float* zr = z   + (size_t)r  * E_DIM;
    float* zq = out + OFF_ZQ + (size_t)r * E_DIM;

    float local = 0.f;
    #pragma unroll
    for (int c = lane; c < E_DIM; c += 32) {
        float ev = e[c], zv = zr[c];
        zq[c] = ev;                      // z_q_st == z_q numerically
        float d = ev - zv;
        local += d * d;
    }
    #pragma unroll
    for (int off = 16; off > 0; off >>= 1) local += __shfl_xor(local, off, 32);

    if (lane == 0) {
        atomicAdd(out + OFF_LOSS, local * ((1.0f + BETA) / (float)(NROWS * E_DIM)));
        atomicAdd(counts + id, 1.0f);
        out[OFF_ENC + (size_t)r * N_E + id] = 1.0f;
        out[OFF_IDX + r] = (float)id;
    }
}

// ---------------- kernel 5: perplexity --------------------------------------
__global__ void vq_perp_kernel(const float* __restrict__ counts, float* __restrict__ out) {
    __shared__ float red[1024];
    int t = threadIdx.x;
    float c = counts[t] * (1.0f / (float)NROWS);
    red[t] = c * logf(c + 1e-10f);
    __syncthreads();
    for (int s = 512; s > 0; s >>= 1) {
        if (t < s) red[t] += red[t + s];
        __syncthreads();
    }
    if (t == 0) out[OFF_PERP] = expf(-red[0]);
}

extern "C" void kernel_launch(void* const* d_in, const int* in_sizes, int n_in,
                              void* d_out, int out_size, void* d_ws, size_t ws_size,
                              hipStream_t stream) {
    const float* z   = (const float*)d_in[0];   // (32,512,128)
    const float* emb = (const float*)d_in[1];   // (1024,128)
    float* out = (float*)d_out;

    float* e2     = (float*)d_ws;               // 1024 floats
    float* counts = e2 + N_E;                   // 1024 floats
    int*   idxb   = (int*)(counts + N_E);       // 16384 ints

    vq_zero_kernel<<<2048, 256, 0, stream>>>(out, counts);
    vq_e2_kernel<<<(N_E + 255) / 256, 256, 0, stream>>>(emb, e2);
    vq_argmin_wmma<<<128, 256, 0, stream>>>(z, emb, e2, idxb);   // 1024 waves
    vq_gather_kernel<<<2048, 256, 0, stream>>>(z, emb, idxb, out, counts); // 16384 waves
    vq_perp_kernel<<<1, 1024, 0, stream>>>(counts, out);
}